// GNNModelClassification_72310069396108
// MI455X (gfx1250) — compile-verified
//
#include <hip/hip_runtime.h>
#include <hip/hip_bf16.h>
#include <cmath>

typedef __attribute__((ext_vector_type(2))) float v2f;
typedef __attribute__((ext_vector_type(8))) float v8f;

// ---------------------------------------------------------------------------
// float atomic max via int ordering trick (exact, deterministic result)
// ---------------------------------------------------------------------------
__device__ __forceinline__ void atomicMaxF(float* addr, float val) {
    if (val >= 0.0f) atomicMax((int*)addr, __float_as_int(val));
    else             atomicMin((unsigned int*)addr, __float_as_uint(val));
}

// ---------------------------------------------------------------------------
// generic fill
// ---------------------------------------------------------------------------
__global__ void fill_f32(float* __restrict__ p, long n, float v) {
    long i = (long)blockIdx.x * blockDim.x + threadIdx.x;
    if (i < n) p[i] = v;
}

// ---------------------------------------------------------------------------
// Fused 4-way node GEMM:  O_j = X @ W_j + b_j  for j in {q,k,v,skip}.
// One wave owns a 16-row tile of X; A fragments loaded ONCE and reused across
// all 4 weight matrices and all N-tiles (4*(D/16)*KB WMMAs per A-load).
// V_WMMA_F32_16X16X4_F32 layout: lane = half*16 + ln;
//   A: row=ln, k = kb*4 + half*2 + {0,1};  B: col=ln, same k;
//   C/D: col=ln, row = r + half*8 (r=0..7).
// Full-tile fast path keeps the store epilogue EXEC-uniform (no saveexec chains).
// ---------------------------------------------------------------------------
template<int K, int D>
__global__ void node_gemm4(const float* __restrict__ X, int N,
                           const float* __restrict__ W0, const float* __restrict__ W1,
                           const float* __restrict__ W2, const float* __restrict__ W3,
                           const float* __restrict__ B0, const float* __restrict__ B1,
                           const float* __restrict__ B2, const float* __restrict__ B3,
                           float* __restrict__ O0, float* __restrict__ O1,
                           float* __restrict__ O2, float* __restrict__ O3)
{
    const int wave = threadIdx.x >> 5;
    const int lane = threadIdx.x & 31;
    const int half = lane >> 4;
    const int ln   = lane & 15;
    const int row0 = (blockIdx.x * (blockDim.x >> 5) + wave) * 16;
    if (row0 >= N) return;  // wave-uniform

    constexpr int KB = (K + 3) / 4;
    const int  m    = row0 + ln;
    const bool mok  = (m < N);
    const bool full = (row0 + 16 <= N);  // wave-uniform

    v2f afrag[KB];
#pragma unroll
    for (int kb = 0; kb < KB; ++kb) {
        const int k0 = kb * 4 + half * 2;
        v2f a;
        a.x = (mok && (k0     < K)) ? X[(size_t)m * K + k0    ] : 0.0f;
        a.y = (mok && (k0 + 1 < K)) ? X[(size_t)m * K + k0 + 1] : 0.0f;
        afrag[kb] = a;
    }

    const float* Ws[4] = {W0, W1, W2, W3};
    const float* Bs[4] = {B0, B1, B2, B3};
    float*       Os[4] = {O0, O1, O2, O3};

#pragma unroll
    for (int which = 0; which < 4; ++which) {
        const float* W  = Ws[which];
        const float* Bi = Bs[which];
        float*       O  = Os[which];
#pragma unroll
        for (int nt = 0; nt < D / 16; ++nt) {
            const int col = nt * 16 + ln;
            v8f acc = {0.f, 0.f, 0.f, 0.f, 0.f, 0.f, 0.f, 0.f};
#pragma unroll
            for (int kb = 0; kb < KB; ++kb) {
                const int k0 = kb * 4 + half * 2;
                v2f b;
                b.x = (k0     < K) ? W[(size_t)(k0    ) * D + col] : 0.0f;
                b.y = (k0 + 1 < K) ? W[(size_t)(k0 + 1) * D + col] : 0.0f;
                acc = __builtin_amdgcn_wmma_f32_16x16x4_f32(false, afrag[kb], false, b,
                                                            (short)0, acc, false, false);
            }
            const float bias = Bi[col];
            if (full) {
#pragma unroll
                for (int r = 0; r < 8; ++r) {
                    const int row = row0 + half * 8 + r;
                    O[(size_t)row * D + col] = acc[r] + bias;
                }
            } else {
#pragma unroll
                for (int r = 0; r < 8; ++r) {
                    const int row = row0 + half * 8 + r;
                    if (row < N) O[(size_t)row * D + col] = acc[r] + bias;
                }
            }
        }
    }
}

// ---------------------------------------------------------------------------
// Edge pass 1: score[e] = scale * dot(q[dst], k[src]); segment max into mrow[dst]
// ---------------------------------------------------------------------------
template<int D>
__global__ void edge_score_max(const long long* __restrict__ src, const long long* __restrict__ dst,
                               int E, const float* __restrict__ q, const float* __restrict__ k,
                               float* __restrict__ score, float* __restrict__ mrow, float scale)
{
    int e = blockIdx.x * blockDim.x + threadIdx.x;
    if (e >= E) return;
    const int s = (int)src[e], d = (int)dst[e];
    const float4* qp = (const float4*)(q + (size_t)d * D);
    const float4* kp = (const float4*)(k + (size_t)s * D);
    float acc = 0.f;
#pragma unroll
    for (int i = 0; i < D / 4; ++i) {
        float4 a = qp[i], b = kp[i];
        acc += a.x * b.x + a.y * b.y + a.z * b.z + a.w * b.w;
    }
    acc *= scale;
    score[e] = acc;
    atomicMaxF(mrow + d, acc);
}

// ---------------------------------------------------------------------------
// Edge pass 2: score[e] <- exp(score - max[dst]); segment sum into den[dst]
// ---------------------------------------------------------------------------
__global__ void edge_exp(const long long* __restrict__ dst, int E,
                         float* __restrict__ score, const float* __restrict__ mrow,
                         float* __restrict__ den)
{
    int e = blockIdx.x * blockDim.x + threadIdx.x;
    if (e >= E) return;
    const int d = (int)dst[e];
    float v = __expf(score[e] - mrow[d]);
    score[e] = v;
    atomicAdd(den + d, v);
}

// ---------------------------------------------------------------------------
// Edge pass 3: agg[dst] += (score[e]/den[dst]) * v[src]
// ---------------------------------------------------------------------------
template<int D>
__global__ void edge_agg(const long long* __restrict__ src, const long long* __restrict__ dst, int E,
                         const float* __restrict__ escore, const float* __restrict__ den,
                         const float* __restrict__ vmat, float* __restrict__ agg)
{
    int e = blockIdx.x * blockDim.x + threadIdx.x;
    if (e >= E) return;
    const int s = (int)src[e], d = (int)dst[e];
    const float w = escore[e] / den[d];
    const float4* vp = (const float4*)(vmat + (size_t)s * D);
    float* ap = agg + (size_t)d * D;
#pragma unroll
    for (int i = 0; i < D / 4; ++i) {
        float4 t = vp[i];
        atomicAdd(ap + 4 * i + 0, w * t.x);
        atomicAdd(ap + 4 * i + 1, w * t.y);
        atomicAdd(ap + 4 * i + 2, w * t.z);
        atomicAdd(ap + 4 * i + 3, w * t.w);
    }
}

// ---------------------------------------------------------------------------
// h = relu(agg + skip)
// ---------------------------------------------------------------------------
__global__ void node_relu_skip(const float* __restrict__ agg, const float* __restrict__ sk,
                               float* __restrict__ o, long n)
{
    long i = (long)blockIdx.x * blockDim.x + threadIdx.x;
    if (i < n) o[i] = fmaxf(agg[i] + sk[i], 0.0f);
}

// ---------------------------------------------------------------------------
// global mean pool: accumulate sums/counts per graph
// ---------------------------------------------------------------------------
__global__ void pool_accum(const float* __restrict__ h, const long long* __restrict__ batch,
                           int N, float* __restrict__ sums, float* __restrict__ cnts)
{
    int n = blockIdx.x * blockDim.x + threadIdx.x;
    if (n >= N) return;
    const int b = (int)batch[n];
    atomicAdd(cnts + b, 1.0f);
    const float4* hp = (const float4*)(h + (size_t)n * 32);
    float* sp = sums + (size_t)b * 32;
#pragma unroll
    for (int i = 0; i < 8; ++i) {
        float4 t = hp[i];
        atomicAdd(sp + 4 * i + 0, t.x);
        atomicAdd(sp + 4 * i + 1, t.y);
        atomicAdd(sp + 4 * i + 2, t.z);
        atomicAdd(sp + 4 * i + 3, t.w);
    }
}

__global__ void pool_div(float* __restrict__ sums, const float* __restrict__ cnts, int Bn)
{
    int i = blockIdx.x * blockDim.x + threadIdx.x;
    if (i >= Bn * 32) return;
    sums[i] /= fmaxf(cnts[i >> 5], 1.0f);
}

// ---------------------------------------------------------------------------
// head fc1: z = relu(pooled[1000x32] @ Wf1[32x64] + bf1) via f32 WMMA
// ---------------------------------------------------------------------------
__global__ void head_fc1_wmma(const float* __restrict__ P, int Bn,
                              const float* __restrict__ W, const float* __restrict__ Bi,
                              float* __restrict__ Z)
{
    const int lane = threadIdx.x & 31;
    const int half = lane >> 4;
    const int ln   = lane & 15;
    const int row0 = blockIdx.x * 16;
    constexpr int K = 32, D = 64;
    const int  m    = row0 + ln;
    const bool mok  = (m < Bn);
    const bool full = (row0 + 16 <= Bn);  // wave-uniform

    v2f afrag[8];
#pragma unroll
    for (int kb = 0; kb < 8; ++kb) {
        const int k0 = kb * 4 + half * 2;
        v2f a;
        a.x = mok ? P[(size_t)m * K + k0    ] : 0.0f;
        a.y = mok ? P[(size_t)m * K + k0 + 1] : 0.0f;
        afrag[kb] = a;
    }
#pragma unroll
    for (int nt = 0; nt < 4; ++nt) {
        const int col = nt * 16 + ln;
        v8f acc = {0.f, 0.f, 0.f, 0.f, 0.f, 0.f, 0.f, 0.f};
#pragma unroll
        for (int kb = 0; kb < 8; ++kb) {
            const int k0 = kb * 4 + half * 2;
            v2f b;
            b.x = W[(size_t)(k0    ) * D + col];
            b.y = W[(size_t)(k0 + 1) * D + col];
            acc = __builtin_amdgcn_wmma_f32_16x16x4_f32(false, afrag[kb], false, b,
                                                        (short)0, acc, false, false);
        }
        const float bias = Bi[col];
        if (full) {
#pragma unroll
            for (int r = 0; r < 8; ++r) {
                const int row = row0 + half * 8 + r;
                Z[(size_t)row * D + col] = fmaxf(acc[r] + bias, 0.0f);
            }
        } else {
#pragma unroll
            for (int r = 0; r < 8; ++r) {
                const int row = row0 + half * 8 + r;
                if (row < Bn) Z[(size_t)row * D + col] = fmaxf(acc[r] + bias, 0.0f);
            }
        }
    }
}

// ---------------------------------------------------------------------------
// head fc2 + sigmoid: out[b][j] = sigmoid(z[b]@Wf2[:,j] + bf2[j])
// ---------------------------------------------------------------------------
__global__ void head_fc2(const float* __restrict__ Z, const float* __restrict__ W,
                         const float* __restrict__ Bi, float* __restrict__ out, int Bn)
{
    int i = blockIdx.x * blockDim.x + threadIdx.x;
    if (i >= Bn * 2) return;
    const int b = i >> 1, j = i & 1;
    float acc = Bi[j];
    const float* z = Z + (size_t)b * 64;
#pragma unroll
    for (int c = 0; c < 64; ++c) acc += z[c] * W[c * 2 + j];
    out[i] = 1.0f / (1.0f + __expf(-acc));
}

// ---------------------------------------------------------------------------
extern "C" void kernel_launch(void* const* d_in, const int* in_sizes, int n_in,
                              void* d_out, int out_size, void* d_ws, size_t ws_size,
                              hipStream_t stream)
{
    const float*     x     = (const float*)d_in[0];
    const long long* ei    = (const long long*)d_in[1];
    const long long* batch = (const long long*)d_in[2];
    const float *Wq1 = (const float*)d_in[3],  *bq1 = (const float*)d_in[4];
    const float *Wk1 = (const float*)d_in[5],  *bk1 = (const float*)d_in[6];
    const float *Wv1 = (const float*)d_in[7],  *bv1 = (const float*)d_in[8];
    const float *Ws1 = (const float*)d_in[9],  *bs1 = (const float*)d_in[10];
    const float *Wq2 = (const float*)d_in[11], *bq2 = (const float*)d_in[12];
    const float *Wk2 = (const float*)d_in[13], *bk2 = (const float*)d_in[14];
    const float *Wv2 = (const float*)d_in[15], *bv2 = (const float*)d_in[16];
    const float *Ws2 = (const float*)d_in[17], *bs2 = (const float*)d_in[18];
    const float *Wf1 = (const float*)d_in[19], *bf1 = (const float*)d_in[20];
    const float *Wf2 = (const float*)d_in[21], *bf2 = (const float*)d_in[22];
    float* out = (float*)d_out;

    const int N  = in_sizes[0] / 9;
    const int E  = in_sizes[1] / 2;
    const int Bn = 1000;
    const long long* src = ei;
    const long long* dst = ei + E;

    // ---- workspace carve (floats) ----
    float* ws = (float*)d_ws;
    size_t off = 0;
    auto carve = [&](size_t n) { float* p = ws + off; off += n; return p; };
    float* q    = carve((size_t)N * 32);
    float* kbuf = carve((size_t)N * 32);
    float* vbuf = carve((size_t)N * 32);
    float* sk   = carve((size_t)N * 32);
    float* agg  = carve((size_t)N * 32);
    float* h    = carve((size_t)N * 32);
    float* scr  = carve((size_t)E);
    float* mrow = carve((size_t)N);
    float* den  = carve((size_t)N);
    float* sums = carve((size_t)Bn * 32);
    float* cnts = carve((size_t)Bn);
    float* z1   = carve((size_t)Bn * 64);
    (void)ws_size; (void)n_in; (void)out_size;

    auto cdiv = [](long a, long b) { return (int)((a + b - 1) / b); };
    dim3 blk(256);

    // ================= layer 1 (9 -> 16) =================
    {
        node_gemm4<9, 16><<<dim3(cdiv(N, 128)), blk, 0, stream>>>(
            x, N, Wq1, Wk1, Wv1, Ws1, bq1, bk1, bv1, bs1, q, kbuf, vbuf, sk);
        fill_f32<<<cdiv(N, 256), blk, 0, stream>>>(mrow, N, -INFINITY);
        fill_f32<<<cdiv(N, 256), blk, 0, stream>>>(den, N, 0.0f);
        fill_f32<<<cdiv((long)N * 16, 256), blk, 0, stream>>>(agg, (long)N * 16, 0.0f);
        edge_score_max<16><<<cdiv(E, 256), blk, 0, stream>>>(src, dst, E, q, kbuf, scr, mrow, 0.25f);
        edge_exp<<<cdiv(E, 256), blk, 0, stream>>>(dst, E, scr, mrow, den);
        edge_agg<16><<<cdiv(E, 256), blk, 0, stream>>>(src, dst, E, scr, den, vbuf, agg);
        node_relu_skip<<<cdiv((long)N * 16, 256), blk, 0, stream>>>(agg, sk, h, (long)N * 16);
    }

    // ================= layer 2 (16 -> 32) =================
    {
        node_gemm4<16, 32><<<dim3(cdiv(N, 128)), blk, 0, stream>>>(
            h, N, Wq2, Wk2, Wv2, Ws2, bq2, bk2, bv2, bs2, q, kbuf, vbuf, sk);
        fill_f32<<<cdiv(N, 256), blk, 0, stream>>>(mrow, N, -INFINITY);
        fill_f32<<<cdiv(N, 256), blk, 0, stream>>>(den, N, 0.0f);
        fill_f32<<<cdiv((long)N * 32, 256), blk, 0, stream>>>(agg, (long)N * 32, 0.0f);
        edge_score_max<32><<<cdiv(E, 256), blk, 0, stream>>>(src, dst, E, q, kbuf, scr, mrow,
                                                             0.17677669529663688f);
        edge_exp<<<cdiv(E, 256), blk, 0, stream>>>(dst, E, scr, mrow, den);
        edge_agg<32><<<cdiv(E, 256), blk, 0, stream>>>(src, dst, E, scr, den, vbuf, agg);
        node_relu_skip<<<cdiv((long)N * 32, 256), blk, 0, stream>>>(agg, sk, h, (long)N * 32);
    }

    // ================= global mean pool =================
    fill_f32<<<cdiv((long)Bn * 32, 256), blk, 0, stream>>>(sums, (long)Bn * 32, 0.0f);
    fill_f32<<<cdiv(Bn, 256), blk, 0, stream>>>(cnts, Bn, 0.0f);
    pool_accum<<<cdiv(N, 256), blk, 0, stream>>>(h, batch, N, sums, cnts);
    pool_div<<<cdiv((long)Bn * 32, 256), blk, 0, stream>>>(sums, cnts, Bn);

    // ================= head =================
    head_fc1_wmma<<<dim3(cdiv(Bn, 16)), dim3(32), 0, stream>>>(sums, Bn, Wf1, bf1, z1);
    head_fc2<<<cdiv((long)Bn * 2, 256), blk, 0, stream>>>(z1, Wf2, bf2, out, Bn);
}